// SSD_77094662963263
// MI455X (gfx1250) — compile-verified
//
#include <hip/hip_runtime.h>

// ---------------------------------------------------------------------------
// BlazeFace-SSD forward for MI455X (gfx1250).
// Pointwise (1x1) convs = fp32 WMMA GEMMs (v_wmma_f32_16x16x4_f32) with the
// A (weight) fragment reused across 4 pixel tiles per wave (16x64 per wave),
// BN / ReLU / residual fused in the epilogue. Depthwise 5x5 / stem / maxpool /
// 3x3 heads are bandwidth-bound VALU kernels.
// ---------------------------------------------------------------------------

typedef __attribute__((ext_vector_type(2))) float v2f;
typedef __attribute__((ext_vector_type(8))) float v8f;

#define BN_EPS 1e-5f

// ---------------- stem: 3x3 s2 p1, 3->24, + bias + BN + ReLU ---------------
__global__ void stem_kernel(const float* __restrict__ x, const float* __restrict__ w,
                            const float* __restrict__ bias,
                            const float* __restrict__ g, const float* __restrict__ bb,
                            const float* __restrict__ mm, const float* __restrict__ vv,
                            float* __restrict__ out) {
  const int HO = 128, HI = 256, CO = 24, CI = 3;
  int idx = blockIdx.x * blockDim.x + threadIdx.x;
  int wo = idx % HO; int t = idx / HO;
  int ho = t % HO;  t /= HO;
  int co = t % CO;  int b = t / CO;
  float acc = 0.f;
  for (int ci = 0; ci < CI; ++ci) {
    const float* ic = x + ((size_t)b * CI + ci) * HI * HI;
    const float* wc = w + (co * CI + ci) * 9;
#pragma unroll
    for (int ky = 0; ky < 3; ++ky) {
      int iy = ho * 2 - 1 + ky;
      if ((unsigned)iy >= (unsigned)HI) continue;
#pragma unroll
      for (int kx = 0; kx < 3; ++kx) {
        int ix = wo * 2 - 1 + kx;
        if ((unsigned)ix >= (unsigned)HI) continue;
        acc += ic[(size_t)iy * HI + ix] * wc[ky * 3 + kx];
      }
    }
  }
  float sc = g[co] * rsqrtf(vv[co] + BN_EPS);
  float sh = (bias[co] - mm[co]) * sc + bb[co];
  out[((size_t)b * CO + co) * HO * HO + (size_t)ho * HO + wo] =
      fmaxf(acc * sc + sh, 0.f);
}

// ------------- depthwise 5x5 pad 2, stride 1|2, + bias + BN ----------------
__global__ void dw5x5_kernel(const float* __restrict__ in, const float* __restrict__ w,
                             const float* __restrict__ bias,
                             const float* __restrict__ g, const float* __restrict__ bb,
                             const float* __restrict__ mm, const float* __restrict__ vv,
                             float* __restrict__ out, int C, int HI, int stride) {
  int HO = (stride == 2) ? (HI >> 1) : HI;
  int idx = blockIdx.x * blockDim.x + threadIdx.x;
  int wo = idx % HO; int t = idx / HO;
  int ho = t % HO;  t /= HO;
  int c  = t % C;   int b = t / C;
  const float* ip = in + ((size_t)b * C + c) * HI * HI;
  const float* wp = w + c * 25;
  float acc = 0.f;
  int y0 = ho * stride - 2, x0 = wo * stride - 2;
#pragma unroll
  for (int ky = 0; ky < 5; ++ky) {
    int iy = y0 + ky;
    if ((unsigned)iy >= (unsigned)HI) continue;
#pragma unroll
    for (int kx = 0; kx < 5; ++kx) {
      int ix = x0 + kx;
      if ((unsigned)ix >= (unsigned)HI) continue;
      acc += ip[(size_t)iy * HI + ix] * wp[ky * 5 + kx];
    }
  }
  float sc = g[c] * rsqrtf(vv[c] + BN_EPS);
  float sh = (bias[c] - mm[c]) * sc + bb[c];
  out[((size_t)b * C + c) * HO * HO + (size_t)ho * HO + wo] = acc * sc + sh;
}

// --------------------------- 2x2 stride-2 maxpool --------------------------
__global__ void maxpool2_kernel(const float* __restrict__ in, float* __restrict__ out,
                                int C, int HI) {
  int HO = HI >> 1;
  int idx = blockIdx.x * blockDim.x + threadIdx.x;
  int wo = idx % HO; int t = idx / HO;
  int ho = t % HO;  t /= HO;
  int c  = t % C;   int b = t / C;
  const float* ip = in + ((size_t)b * C + c) * HI * HI + (size_t)(ho * 2) * HI + wo * 2;
  float m = fmaxf(fmaxf(ip[0], ip[1]), fmaxf(ip[HI], ip[HI + 1]));
  out[((size_t)b * C + c) * HO * HO + (size_t)ho * HO + wo] = m;
}

// ---- pointwise 1x1 conv as WMMA GEMM: Out[Cout x HW] = W[Cout x Cin]*In ---
// Fused: +conv-bias, BN, optional residual (channel-padded), ReLU.
// One wave computes a 16(M) x 16*NT(N) tile. Per K-step: one A (weight)
// fragment load shared by NT back-to-back v_wmma_f32_16x16x4_f32 ops
// (A-reuse cuts weight traffic NT x and amortizes address math).
// Grid: x = npix/(128*NT), y = M tiles, z = batch. npix is a multiple of
// 128*NT for every layer it is launched on -> no divergence before/during
// the WMMA loop (EXEC all-ones as the ISA requires).
template <int NT>
__global__ void pw_wmma_kernel(const float* __restrict__ in, const float* __restrict__ w,
                               const float* __restrict__ bias,
                               const float* __restrict__ g, const float* __restrict__ bb,
                               const float* __restrict__ mm, const float* __restrict__ vv,
                               const float* __restrict__ res, int res_ch,
                               float* __restrict__ out, int Cin, int Cout, int npix) {
  int wave = threadIdx.x >> 5;
  int lane = threadIdx.x & 31;
  int l  = lane & 15;      // position within half-wave
  int kh = lane >> 4;      // half-wave select (K/M split per ISA layouts)
  int n0 = (blockIdx.x * 8 + wave) * (16 * NT);
  int m0 = blockIdx.y * 16;
  int b  = blockIdx.z;
  const float* inb = in + (size_t)b * Cin * npix;

  int ma = m0 + l; ma = (ma < Cout) ? ma : (Cout - 1);  // clamp partial M tile
  const float* wrow = w + (size_t)ma * Cin + 2 * kh;

  v8f acc[NT];
#pragma unroll
  for (int t = 0; t < NT; ++t) acc[t] = (v8f){};

  for (int k = 0; k < Cin; k += 4) {           // Cin is always a multiple of 4
    // A (16x4): lanes 0-15 hold M=lane, K={0,1}; lanes 16-31 K={2,3}
    v2f av;
    av.x = wrow[k];
    av.y = wrow[k + 1];
    // B (4x16): VGPR0 rows K={0,2}, VGPR1 rows K={1,3}, N striped over lanes
    const float* bp = inb + (size_t)(k + 2 * kh) * npix + n0 + l;
    v2f bv[NT];
#pragma unroll
    for (int t = 0; t < NT; ++t) {
      bv[t].x = bp[t * 16];
      bv[t].y = bp[t * 16 + npix];
    }
#pragma unroll
    for (int t = 0; t < NT; ++t) {
      acc[t] = __builtin_amdgcn_wmma_f32_16x16x4_f32(
          /*neg_a=*/false, av, /*neg_b=*/false, bv[t],
          /*c_mod=*/(short)0, acc[t], /*reuse_a=*/false, /*reuse_b=*/false);
    }
  }

  // Epilogue (divergence here is fine; WMMA is done).
  float* outb = out + (size_t)b * Cout * npix;
#pragma unroll
  for (int r = 0; r < 8; ++r) {
    int m = m0 + r + 8 * kh;   // C/D layout: VGPR r -> rows m0+r / m0+r+8
    if (m < Cout) {
      float sc = g[m] * rsqrtf(vv[m] + BN_EPS);
      float sh = (bias[m] - mm[m]) * sc + bb[m];
      const float* rrow = res ? (res + ((size_t)b * res_ch + m) * npix + n0 + l) : nullptr;
      float* orow = outb + (size_t)m * npix + n0 + l;
#pragma unroll
      for (int t = 0; t < NT; ++t) {
        float o = acc[t][r] * sc + sh;
        if (res && m < res_ch) o += rrow[t * 16];
        orow[t * 16] = fmaxf(o, 0.f);  // both pw stages end in ReLU
      }
    }
  }
}

// ------ 3x3 p1 head conv, Cin=96, conv+bias only, writes NHWC into d_out ---
__global__ void head3x3_kernel(const float* __restrict__ in, const float* __restrict__ w,
                               const float* __restrict__ bias, float* __restrict__ out,
                               int CO, int H,
                               unsigned long long bstride, unsigned long long baseoff) {
  const int CI = 96;
  int idx = blockIdx.x * blockDim.x + threadIdx.x;
  int wo = idx % H; int t = idx / H;
  int ho = t % H;  t /= H;
  int co = t % CO; int b = t / CO;
  float acc = bias[co];
  const float* ib = in + (size_t)b * CI * H * H;
  const float* wb = w + (size_t)co * CI * 9;
  for (int ci = 0; ci < CI; ++ci) {
    const float* ic = ib + (size_t)ci * H * H;
    const float* wc = wb + ci * 9;
#pragma unroll
    for (int ky = 0; ky < 3; ++ky) {
      int iy = ho - 1 + ky;
      if ((unsigned)iy >= (unsigned)H) continue;
#pragma unroll
      for (int kx = 0; kx < 3; ++kx) {
        int ix = wo - 1 + kx;
        if ((unsigned)ix >= (unsigned)H) continue;
        acc += ic[(size_t)iy * H + ix] * wc[ky * 3 + kx];
      }
    }
  }
  out[baseoff + (size_t)b * bstride + ((size_t)ho * H + wo) * CO + co] = acc;
}

// ---------------------------------------------------------------------------
// Host orchestration.
// Flattened input order (setup_inputs dict order, nested dicts/lists in order):
//  0: x
//  1: stem_w  2: stem_b  3..6: stem_bn {g,b,m,v}
//  7 + blk*24: per block: dw1w,dw1b,bn1{g,b,m,v}, pw1w,pw1b,bn2{g,b,m,v},
//                         dw2w,dw2b,bn3{g,b,m,v}, pw2w,pw2b,bn4{g,b,m,v}
//  271..274: loc[0].{w,b}, loc[1].{w,b}   275..278: conf[0..1].{w,b}
//  279: dbox
// Output: loc [64*30720] | conf [64*15360] | dbox [30720], all fp32.
// ---------------------------------------------------------------------------
extern "C" void kernel_launch(void* const* d_in, const int* in_sizes, int n_in,
                              void* d_out, int out_size, void* d_ws, size_t ws_size,
                              hipStream_t stream) {
  (void)in_sizes; (void)n_in; (void)out_size; (void)ws_size;
  auto F = [&](int i) { return (const float*)d_in[i]; };
  float* outp = (float*)d_out;

  // 4 activation buffers of 64*24*128*128 floats (100.7 MB each, ~403 MB total)
  const size_t S = (size_t)64 * 24 * 128 * 128;
  float* ws = (float*)d_ws;
  float* bufX  = ws;
  float* bufT1 = ws + S;
  float* bufT2 = ws + 2 * S;
  float* bufP  = ws + 3 * S;

  const int B = 64;

  auto launch_pw = [&](const float* in, int base, const float* res, int res_ch,
                       float* out, int Cin, int Cout, int npix) {
    if (npix % 512 == 0) {
      pw_wmma_kernel<4><<<dim3((unsigned)(npix / 512), (unsigned)((Cout + 15) / 16), B),
                          dim3(256), 0, stream>>>(
          in, F(base), F(base + 1), F(base + 2), F(base + 3), F(base + 4), F(base + 5),
          res, res_ch, out, Cin, Cout, npix);
    } else {
      pw_wmma_kernel<1><<<dim3((unsigned)(npix / 128), (unsigned)((Cout + 15) / 16), B),
                          dim3(256), 0, stream>>>(
          in, F(base), F(base + 1), F(base + 2), F(base + 3), F(base + 4), F(base + 5),
          res, res_ch, out, Cin, Cout, npix);
    }
  };

  // stem
  {
    size_t tot = (size_t)B * 24 * 128 * 128;
    stem_kernel<<<dim3((unsigned)(tot / 256)), dim3(256), 0, stream>>>(
        F(0), F(1), F(2), F(3), F(4), F(5), F(6), bufX);
  }

  static const int CFG[11][4] = {
      {24, 24, 24, 1}, {24, 24, 24, 1}, {24, 48, 48, 2}, {48, 48, 48, 1},
      {48, 48, 48, 1}, {48, 24, 96, 2}, {96, 24, 96, 1}, {96, 24, 96, 1},
      {96, 24, 96, 2}, {96, 24, 96, 1}, {96, 24, 96, 1}};

  int H = 128;
  for (int blk = 0; blk < 11; ++blk) {
    int cin = CFG[blk][0], o1 = CFG[blk][1], o2 = CFG[blk][2], s = CFG[blk][3];
    int HO = (s == 2) ? H / 2 : H;
    int npix = HO * HO;
    int base = 7 + blk * 24;

    // dw1 + bn1  (applies stride)
    dw5x5_kernel<<<dim3((unsigned)(((size_t)B * cin * npix) / 256)), dim3(256), 0, stream>>>(
        bufX, F(base + 0), F(base + 1), F(base + 2), F(base + 3), F(base + 4), F(base + 5),
        bufT1, cin, H, s);

    // pw1 + bn2 + relu  (WMMA)
    launch_pw(bufT1, base + 6, nullptr, 0, bufT2, cin, o1, npix);

    // dw2 + bn3  (stride 1)
    dw5x5_kernel<<<dim3((unsigned)(((size_t)B * o1 * npix) / 256)), dim3(256), 0, stream>>>(
        bufT2, F(base + 12), F(base + 13), F(base + 14), F(base + 15), F(base + 16), F(base + 17),
        bufT1, o1, HO, 1);

    // shortcut (maxpool if strided)
    const float* res = bufX;
    if (s == 2) {
      maxpool2_kernel<<<dim3((unsigned)(((size_t)B * cin * npix) / 256)), dim3(256), 0, stream>>>(
          bufX, bufP, cin, H);
      res = bufP;
    }

    // pw2 + bn4 + residual(channel-padded) + relu  (WMMA)
    launch_pw(bufT1, base + 18, res, cin, bufT2, o1, o2, npix);

    { float* t = bufX; bufX = bufT2; bufT2 = t; }  // output becomes next input
    H = HO;

    if (blk == 7) {  // source0: 96ch @ 32x32
      head3x3_kernel<<<dim3((unsigned)(((size_t)B * 24 * npix) / 256)), dim3(256), 0, stream>>>(
          bufX, F(271), F(272), outp, 24, 32, 30720ull, 0ull);
      head3x3_kernel<<<dim3((unsigned)(((size_t)B * 12 * npix) / 256)), dim3(256), 0, stream>>>(
          bufX, F(275), F(276), outp, 12, 32, 15360ull, 1966080ull);
    }
    if (blk == 10) {  // source1: 96ch @ 16x16
      head3x3_kernel<<<dim3((unsigned)(((size_t)B * 24 * npix) / 256)), dim3(256), 0, stream>>>(
          bufX, F(273), F(274), outp, 24, 16, 30720ull, 24576ull);
      head3x3_kernel<<<dim3((unsigned)(((size_t)B * 12 * npix) / 256)), dim3(256), 0, stream>>>(
          bufX, F(277), F(278), outp, 12, 16, 15360ull, 1966080ull + 12288ull);
    }
  }

  // dbox passthrough: loc(64*30720) + conf(64*15360) = 2,949,120 floats offset
  hipMemcpyAsync(outp + 2949120, d_in[279], 30720 * sizeof(float),
                 hipMemcpyDeviceToDevice, stream);
}